// PaiConvSmall_63204738728502
// MI455X (gfx1250) — compile-verified
//
#include <hip/hip_runtime.h>
#include <hip/hip_bf16.h>

typedef float v2f __attribute__((ext_vector_type(2)));
typedef float v8f __attribute__((ext_vector_type(8)));

#define NPTS   50000
#define BATCH  4
#define FCH    64        // IN_C
#define KNB    16        // neighbors
#define OUTC   64
#define NBASES 8
#define FANIN  1024      // KNB * FCH
#define ROWS_PER_BLOCK 16

// Branchless ELU: v_mul (log2e) + v_exp_f32 (TRANS, co-executes with WMMA) + v_cndmask
__device__ __forceinline__ float elu1(float x) {
    return x > 0.0f ? x : __expf(x) - 1.0f;
}

// XOR swizzle: keeps bits [1:0] (b64 alignment) but spreads rows across banks
__device__ __forceinline__ int swz(int row, int q) {
    return q ^ ((row & 15) << 2);
}

__global__ __launch_bounds__(256)
void paiconv_fused_kernel(const float* __restrict__ x,
                          const float* __restrict__ v,
                          const float* __restrict__ adjw,
                          const float* __restrict__ W,
                          const float* __restrict__ bias,
                          const int*   __restrict__ nbidx,
                          float*       __restrict__ out)
{
    __shared__ float ylds[ROWS_PER_BLOCK * FANIN];   // 64 KB, swizzled

    const int tid  = threadIdx.x;
    const int lane = tid & 31;
    const int wave = tid >> 5;
    const int fl   = lane & 15;            // N-index / column within 16
    const int hi   = (lane >> 4) << 1;     // K offset: 0 (lanes 0-15) or 2 (16-31)
    const int tof  = (lane >> 4) << 3;     // M offset in C tiles: 0 or 8
    const int blockBase = blockIdx.x * ROWS_PER_BLOCK;

    // ================= Phase 1: y = ELU(adj^T @ gather(x)) =================
    // wave handles rows 2w, 2w+1 of the block; 16 f32 WMMAs per row.
    for (int rr = 0; rr < 2; ++rr) {
        const int r = wave * 2 + rr;            // row within block
        const int p = blockBase + r;            // global row = b*N + n
        const int b = p / NPTS;
        const int n = p - b * NPTS;

        // v[n, 0..7] (wave-uniform; hardware broadcasts)
        float vv[NBASES];
        #pragma unroll
        for (int s = 0; s < NBASES; ++s) vv[s] = v[n * NBASES + s];

        // A fragments: A[t,k] = adj[k,t] = sum_s v[s]*adjw[s,k,t]
        // lane: t = fl ; component j at step ks: k = ks*4 + hi + j
        v2f afrag[4];
        #pragma unroll
        for (int ks = 0; ks < 4; ++ks) {
            const int k0 = ks * 4 + hi;
            float a0 = 0.f, a1 = 0.f;
            #pragma unroll
            for (int s = 0; s < NBASES; ++s) {
                a0 += vv[s] * adjw[s * 256 + (k0 + 0) * 16 + fl];
                a1 += vv[s] * adjw[s * 256 + (k0 + 1) * 16 + fl];
            }
            afrag[ks].x = a0;
            afrag[ks].y = a1;
        }

        // neighbor row ids for this lane's k slots
        const float* xb = x + (size_t)b * NPTS * FCH;
        int nb0[4], nb1[4];
        #pragma unroll
        for (int ks = 0; ks < 4; ++ks) {
            const int k0 = ks * 4 + hi;
            nb0[ks] = nbidx[p * KNB + k0 + 0];
            nb1[ks] = nbidx[p * KNB + k0 + 1];
        }

        #pragma unroll
        for (int ft = 0; ft < 4; ++ft) {
            v8f acc = {};
            #pragma unroll
            for (int ks = 0; ks < 4; ++ks) {
                v2f bfrag;
                bfrag.x = xb[nb0[ks] * FCH + ft * 16 + fl];
                bfrag.y = xb[nb1[ks] * FCH + ft * 16 + fl];
                acc = __builtin_amdgcn_wmma_f32_16x16x4_f32(
                        false, afrag[ks], false, bfrag,
                        (short)0, acc, false, false);
            }
            // ELU + swizzled store: entry (i, lane) -> t = i+tof, f = ft*16+fl
            #pragma unroll
            for (int i = 0; i < 8; ++i) {
                const int t = i + tof;
                const int q = t * FCH + ft * 16 + fl;     // y flat col (t major)
                ylds[r * FANIN + swz(r, q)] = elu1(acc[i]);
            }
        }
    }

    __syncthreads();

    // ============= Phase 2: out = ELU( y(16x1024) @ W^T + b ) =============
    // 8 waves = 4 column tiles x 2 K-halves of 512 (128 WMMA steps each).
    // Manually unrolled x8: batch 8 LDS b64 + 8 global b64 loads, then the
    // 8-WMMA chain, so the loads-in-flight / wait-skid pattern is forced.
    const int ct   = wave & 3;                // channel tile
    const int half = wave >> 2;               // K half
    const int ccol = ct * 16 + fl;            // output channel for this lane
    const float* wrow = W + (size_t)ccol * FANIN;
    const int m = fl;                         // y row for this lane's A frag

    v8f acc = {};
    const int qs0 = half * 128;
    for (int qq = 0; qq < 128; qq += 8) {
        v2f af[8], wf[8];
        #pragma unroll
        for (int u = 0; u < 8; ++u) {
            const int q = (qs0 + qq + u) * 4 + hi;
            af[u] = *(const v2f*)&ylds[m * FANIN + swz(m, q)];  // ds_load_b64
            wf[u] = *(const v2f*)&wrow[q];                      // global_load_b64
        }
        #pragma unroll
        for (int u = 0; u < 8; ++u) {
            acc = __builtin_amdgcn_wmma_f32_16x16x4_f32(
                    false, af[u], false, wf[u], (short)0, acc, false, false);
        }
    }

    __syncthreads();     // all ylds reads complete; safe to reuse for reduction

    if (half == 1) {
        #pragma unroll
        for (int i = 0; i < 8; ++i)
            ylds[(ct * 8 + i) * 32 + lane] = acc[i];
    }
    __syncthreads();

    if (half == 0) {
        const float bb = bias[ccol];
        #pragma unroll
        for (int i = 0; i < 8; ++i) {
            float sum = acc[i] + ylds[(ct * 8 + i) * 32 + lane] + bb;
            float e = elu1(sum);
            const int mrow = i + tof;
            const int p = blockBase + mrow;
            const int n = p % NPTS;
            if (n == NPTS - 1) e = 0.0f;     // zero_pad: last point zeroed
            out[(size_t)p * OUTC + ccol] = e;
        }
    }
}

extern "C" void kernel_launch(void* const* d_in, const int* in_sizes, int n_in,
                              void* d_out, int out_size, void* d_ws, size_t ws_size,
                              hipStream_t stream) {
    const float* x     = (const float*)d_in[0];
    const float* v     = (const float*)d_in[1];
    const float* adjw  = (const float*)d_in[2];
    const float* W     = (const float*)d_in[3];
    const float* bias  = (const float*)d_in[4];
    const int*   nbidx = (const int*)d_in[5];
    float* out = (float*)d_out;

    const int total_rows = BATCH * NPTS;                 // 200000
    const int grid = total_rows / ROWS_PER_BLOCK;        // 12500 (exact)
    paiconv_fused_kernel<<<grid, 256, 0, stream>>>(x, v, adjw, W, bias, nbidx, out);
}